// YOLOXPostprocess_49185965473944
// MI455X (gfx1250) — compile-verified
//
#include <hip/hip_runtime.h>
#include <stdint.h>

// ---------------------------------------------------------------------------
// YOLOX postprocess for MI455X (gfx1250, wave32, WGP w/ 320KB LDS).
//
// Perf reasoning (no runtime available):
//  - Input traffic ~91MB fp32 -> ~4us @ 23.3 TB/s; everything else is tiny,
//    so kernel 1 must be a single coalesced pass (sigmoid(max) fusion, fast
//    v_rcp/v_exp transcendentals so issue slots hide under the HBM stream).
//  - Whole per-image top-k + NMS lives in one workgroup: 128KB LDS bitonic
//    sort (needs CDNA5's 320KB LDS), async global->LDS gather of top boxes
//    (ASYNCcnt path), IoU union-sum via v_wmma_f32_16x16x4_f32 rank-2 outer
//    product, division-free threshold test (inter > thr*union), ballot-packed
//    suppression bitmask, single-wave32 bitmask NMS.
// ---------------------------------------------------------------------------

#define NCLS       80
#define P_TOTAL    8400
#define P_PAD      8448
#define SORT_SZ    16384
#define KTOP       1024
#define NMS_PRE    1000
#define MAXDET     100
#define SCORE_THR  0.01f
#define NMS_THR    0.65f

typedef __attribute__((ext_vector_type(2))) float v2f;
typedef __attribute__((ext_vector_type(8))) float v8f;

__device__ __forceinline__ float sigmoidf(float x) {
    // fast path: v_exp_f32 + v_rcp_f32 (score only feeds threshold/compare/output)
    return __builtin_amdgcn_rcpf(1.0f + __expf(-x));
}

// ---------------------------------------------------------------------------
// Kernel 1: fused sigmoid/argmax/objectness/box-decode. One thread per padded
// point. Class loop is strided by H*W so lanes stay coalesced on x.
// ---------------------------------------------------------------------------
__global__ void decode_kernel(const float* __restrict__ cls0, const float* __restrict__ cls1,
                              const float* __restrict__ cls2,
                              const float* __restrict__ reg0, const float* __restrict__ reg1,
                              const float* __restrict__ reg2,
                              const float* __restrict__ obj0, const float* __restrict__ obj1,
                              const float* __restrict__ obj2,
                              float* __restrict__ boxes_ws, int* __restrict__ labels_ws,
                              float* __restrict__ sortkey, int* __restrict__ sortidx, int nimg)
{
    int gid = blockIdx.x * blockDim.x + threadIdx.x;
    if (gid >= nimg * SORT_SZ) return;
    int n = gid / SORT_SZ, p = gid - n * SORT_SZ;
    sortidx[gid] = p;
    if (p >= P_TOTAL) { sortkey[gid] = -2.0f; return; }   // pad below masked(-1)

    const float *cls, *reg, *obj; int w, s, pl;
    if (p < 6400)      { cls = cls0; reg = reg0; obj = obj0; w = 80; s = 8;  pl = p; }
    else if (p < 8000) { cls = cls1; reg = reg1; obj = obj1; w = 40; s = 16; pl = p - 6400; }
    else               { cls = cls2; reg = reg2; obj = obj2; w = 20; s = 32; pl = p - 8000; }
    int hw = w * w;
    int y = pl / w, x = pl - y * w;

    const float* rb = reg + (size_t)n * 4 * hw + pl;
    float rx = rb[0], ry = rb[(size_t)hw], rw = rb[(size_t)2 * hw], rh = rb[(size_t)3 * hw];
    float ov = obj[(size_t)n * hw + pl];

    const float* cb = cls + (size_t)n * NCLS * hw + pl;
    float best = cb[0]; int bi = 0;
    for (int c = 1; c < NCLS; ++c) {                // argmax keeps first max (strict >)
        float v = cb[(size_t)c * hw];
        if (v > best) { best = v; bi = c; }
    }
    float score = sigmoidf(best) * sigmoidf(ov);    // sigmoid(max cls) == max sigmoid(cls)
    sortkey[gid] = (score >= SCORE_THR) ? score : -1.0f;
    labels_ws[n * P_PAD + p] = bi;

    float fs = (float)s;
    float cx = rx * fs + (float)x * fs, cy = ry * fs + (float)y * fs;
    float hwd = 0.5f * __expf(rw) * fs, hht = 0.5f * __expf(rh) * fs;
    float* bp = boxes_ws + (size_t)(n * P_PAD + p) * 4;
    bp[0] = cx - hwd; bp[1] = cy - hht; bp[2] = cx + hwd; bp[3] = cy + hht;
}

// ---------------------------------------------------------------------------
// Kernel 2: per-image full bitonic sort (descending) of 16384 (key, idx)
// pairs entirely in LDS (128KB of the 320KB WGP LDS). One WG per image.
// ---------------------------------------------------------------------------
__global__ __launch_bounds__(1024) void sort_kernel(const float* __restrict__ sortkey,
                                                    const int* __restrict__ sortidx,
                                                    float* __restrict__ topks,
                                                    int* __restrict__ topki)
{
    __shared__ float ks[SORT_SZ];
    __shared__ int   is[SORT_SZ];
    int n = blockIdx.x, tid = threadIdx.x;
    for (int i = tid; i < SORT_SZ; i += 1024) {
        ks[i] = sortkey[(size_t)n * SORT_SZ + i];
        is[i] = sortidx[(size_t)n * SORT_SZ + i];
    }
    __syncthreads();
    for (unsigned k = 2; k <= SORT_SZ; k <<= 1) {
        for (unsigned j = k >> 1; j > 0; j >>= 1) {
            for (unsigned t = tid; t < SORT_SZ; t += 1024) {
                unsigned ixj = t ^ j;
                if (ixj > t) {
                    bool up = ((t & k) == 0);       // descending overall
                    float a = ks[t], b = ks[ixj];
                    if ((a < b) == up) {
                        ks[t] = b; ks[ixj] = a;
                        int ti = is[t]; is[t] = is[ixj]; is[ixj] = ti;
                    }
                }
            }
            __syncthreads();
        }
    }
    if (tid < KTOP) { topks[n * KTOP + tid] = ks[tid]; topki[n * KTOP + tid] = is[tid]; }
}

// ---------------------------------------------------------------------------
// Kernel 3: fused gather + class-offset + WMMA IoU bitmask + wave32 bitmask
// NMS + compaction. One 1024-thread WG (32 waves) per image. ~156KB LDS.
// ---------------------------------------------------------------------------
__global__ __launch_bounds__(1024) void nms_kernel(const float* __restrict__ boxes_ws,
                                                   const int* __restrict__ labels_ws,
                                                   const float* __restrict__ topks,
                                                   const int* __restrict__ topki,
                                                   float* __restrict__ out, int nimg)
{
    __shared__ float    s_boxes[KTOP * 4];   // 16KB (class-offset boxes)
    __shared__ float    s_area[KTOP];        // 4KB  (also maxabs scratch)
    __shared__ int      s_label[KTOP];       // 4KB
    __shared__ unsigned s_bits[KTOP * 32];   // 128KB IoU>thr bitmask, row-major words
    __shared__ unsigned s_keep[32];          // 1024-bit keep mask
    __shared__ float    s_maxabs;
    __shared__ int      s_total;

    int n = blockIdx.x, tid = threadIdx.x;
    int k = tid;
    float score = topks[n * KTOP + k];
    int   idx   = topki[n * KTOP + k];
    int   lab   = labels_ws[n * P_PAD + idx];
    s_label[k] = lab;
    if (tid < 32) s_keep[tid] = 0u;
    __syncthreads();

    if ((k < NMS_PRE) && (score >= SCORE_THR))
        atomicOr(&s_keep[k >> 5], 1u << (k & 31));     // ds_or builds initial keep

    // --- async global->LDS gather of this thread's top box (ASYNCcnt path) ---
    unsigned lds_addr = (unsigned)(unsigned long long)(void*)&s_boxes[k * 4];
    unsigned long long ga =
        (unsigned long long)(const void*)(boxes_ws + (size_t)(n * P_PAD + idx) * 4);
    asm volatile("global_load_async_to_lds_b128 %0, %1, off"
                 :: "v"(lds_addr), "v"(ga) : "memory");
    asm volatile("s_wait_asynccnt 0" ::: "memory");
    __syncthreads();

    // --- maxabs over first NMS_PRE gathered boxes (tree reduce in s_area) ---
    float lm = 0.f;
    if (k < NMS_PRE)
        for (int c = 0; c < 4; ++c) lm = fmaxf(lm, fabsf(s_boxes[k * 4 + c]));
    s_area[k] = lm;
    __syncthreads();
    for (int o = 512; o > 0; o >>= 1) {
        if (tid < o) s_area[tid] = fmaxf(s_area[tid], s_area[tid + o]);
        __syncthreads();
    }
    if (tid == 0) s_maxabs = s_area[0];
    __syncthreads();

    // --- per-class offset + areas (offset cancels in area) ---
    float offv = (float)lab * (s_maxabs + 1.0f);
    float b0 = s_boxes[k * 4 + 0] + offv, b1 = s_boxes[k * 4 + 1] + offv;
    float b2 = s_boxes[k * 4 + 2] + offv, b3 = s_boxes[k * 4 + 3] + offv;
    s_boxes[k * 4 + 0] = b0; s_boxes[k * 4 + 1] = b1;
    s_boxes[k * 4 + 2] = b2; s_boxes[k * 4 + 3] = b3;
    s_area[k] = (b2 - b0) * (b3 - b1);
    __syncthreads();

    // --- IoU bitmask: 16x16 tiles; union denominator area_i+area_j computed
    //     as a rank-2 outer product on the matrix pipe (v_wmma_f32_16x16x4_f32):
    //     A[m,:] = [area_m, 1, 0, 0],  B[:,n] = [1, area_n, 0, 0].
    //     Threshold test is division-free: inter > thr*(sum - inter + eps),
    //     valid because union > 0 (areas are positive by construction).
    int wave = tid >> 5, lane = tid & 31;
    bool lo = lane < 16;
    int lm16 = lane & 15;
    for (int t = wave; t < (KTOP / 16) * (KTOP / 32); t += 32) {
        int tm = t >> 5, tn = t & 31;
        int m0 = tm * 16, n0 = tn * 32;
        v2f a, bb1, bb2;
        a.x   = lo ? s_area[m0 + lane] : 0.f;          // K=0 (lanes16-31 hold K=2,3 -> 0)
        a.y   = lo ? 1.0f : 0.f;                        // K=1
        bb1.x = lo ? 1.0f : 0.f;
        bb1.y = lo ? s_area[n0 + lane] : 0.f;
        bb2.x = bb1.x;
        bb2.y = lo ? s_area[n0 + 16 + lane] : 0.f;
        v8f cz = {0.f, 0.f, 0.f, 0.f, 0.f, 0.f, 0.f, 0.f};
        v8f d1 = __builtin_amdgcn_wmma_f32_16x16x4_f32(false, a, false, bb1, (short)0, cz, false, false);
        v8f d2 = __builtin_amdgcn_wmma_f32_16x16x4_f32(false, a, false, bb2, (short)0, cz, false, false);

        int nc1 = n0 + lm16, nc2 = nc1 + 16;
        float n1x0 = s_boxes[nc1 * 4 + 0], n1y0 = s_boxes[nc1 * 4 + 1];
        float n1x1 = s_boxes[nc1 * 4 + 2], n1y1 = s_boxes[nc1 * 4 + 3];
        float n2x0 = s_boxes[nc2 * 4 + 0], n2y0 = s_boxes[nc2 * 4 + 1];
        float n2x1 = s_boxes[nc2 * 4 + 2], n2y1 = s_boxes[nc2 * 4 + 3];
        #pragma unroll
        for (int r = 0; r < 8; ++r) {
            int m = m0 + r + (lo ? 0 : 8);             // D layout: lanes16-31 -> M=r+8
            float mx0 = s_boxes[m * 4 + 0], my0 = s_boxes[m * 4 + 1];
            float mx1 = s_boxes[m * 4 + 2], my1 = s_boxes[m * 4 + 3];
            float i1 = fmaxf(fminf(mx1, n1x1) - fmaxf(mx0, n1x0), 0.f) *
                       fmaxf(fminf(my1, n1y1) - fmaxf(my0, n1y0), 0.f);
            float i2 = fmaxf(fminf(mx1, n2x1) - fmaxf(mx0, n2x0), 0.f) *
                       fmaxf(fminf(my1, n2y1) - fmaxf(my0, n2y0), 0.f);
            bool p1 = i1 > NMS_THR * (d1[r] - i1 + 1e-9f);   // iou > thr, no divide
            bool p2 = i2 > NMS_THR * (d2[r] - i2 + 1e-9f);
            unsigned w1 = __builtin_amdgcn_ballot_w32(p1);   // lo16: row r, hi16: row r+8
            unsigned w2 = __builtin_amdgcn_ballot_w32(p2);
            if (lane == 0) {
                s_bits[(m0 + r) * 32 + tn]     = (w1 & 0xFFFFu) | (w2 << 16);
                s_bits[(m0 + r + 8) * 32 + tn] = (w1 >> 16)     | (w2 & 0xFFFF0000u);
            }
        }
    }
    __syncthreads();

    // --- wave32 bitmask NMS: lane w owns keep word w; __shfl broadcasts keep[i] ---
    if (wave == 0) {
        unsigned kw = s_keep[lane];
        for (int i = 0; i < NMS_PRE; ++i) {
            int wi = i >> 5, bi = i & 31;
            unsigned kwi = __shfl(kw, wi, 32);
            if ((kwi >> bi) & 1u) {
                unsigned row = s_bits[i * 32 + lane];
                unsigned gt = (lane > wi) ? 0xFFFFFFFFu
                            : (lane < wi) ? 0u
                            : ((bi == 31) ? 0u : (0xFFFFFFFFu << (bi + 1)));
                kw &= ~(row & gt);
            }
        }
        s_keep[lane] = kw;
        // prefix + total via shuffles
        int base = 0, tot = 0;
        for (int w = 0; w < 32; ++w) {
            int c = __popc(__shfl(kw, w, 32));
            if (w < lane) base += c;
            tot += c;
        }
        if (lane == 0) s_total = tot;
        // compaction (score order preserved): emit up to MAXDET kept detections
        unsigned my = kw; int ro = base;
        while (my) {
            int b = __ffs((int)my) - 1;
            my &= my - 1;
            if (ro < MAXDET) {
                int kk = lane * 32 + b;
                float o3 = (float)s_label[kk] * (s_maxabs + 1.0f);
                float* ob = out + (size_t)(n * MAXDET + ro) * 4;
                ob[0] = s_boxes[kk * 4 + 0] - o3;
                ob[1] = s_boxes[kk * 4 + 1] - o3;
                ob[2] = s_boxes[kk * 4 + 2] - o3;
                ob[3] = s_boxes[kk * 4 + 3] - o3;
                out[(size_t)nimg * 400 + n * MAXDET + ro] = topks[n * KTOP + kk];
                out[(size_t)nimg * 500 + n * MAXDET + ro] = (float)s_label[kk];
            }
            ro++;
        }
    }
    __syncthreads();

    if (tid < MAXDET && tid >= s_total) {              // pad with zeros / label -1
        float* ob = out + (size_t)(n * MAXDET + tid) * 4;
        ob[0] = 0.f; ob[1] = 0.f; ob[2] = 0.f; ob[3] = 0.f;
        out[(size_t)nimg * 400 + n * MAXDET + tid] = 0.f;
        out[(size_t)nimg * 500 + n * MAXDET + tid] = -1.0f;
    }
}

// ---------------------------------------------------------------------------
extern "C" void kernel_launch(void* const* d_in, const int* in_sizes, int n_in,
                              void* d_out, int out_size, void* d_ws, size_t ws_size,
                              hipStream_t stream)
{
    const float* cls0 = (const float*)d_in[0];
    const float* cls1 = (const float*)d_in[1];
    const float* cls2 = (const float*)d_in[2];
    const float* reg0 = (const float*)d_in[3];
    const float* reg1 = (const float*)d_in[4];
    const float* reg2 = (const float*)d_in[5];
    const float* obj0 = (const float*)d_in[6];
    const float* obj1 = (const float*)d_in[7];
    const float* obj2 = (const float*)d_in[8];
    int nimg = in_sizes[0] / (NCLS * 6400);

    // workspace carve-up (~9.9MB for N=32)
    float* boxes_ws = (float*)d_ws;                                  // N*P_PAD*4
    int*   labels_ws = (int*)(boxes_ws + (size_t)nimg * P_PAD * 4);  // N*P_PAD
    float* sortkey  = (float*)(labels_ws + (size_t)nimg * P_PAD);    // N*SORT_SZ
    int*   sortidx  = (int*)(sortkey + (size_t)nimg * SORT_SZ);      // N*SORT_SZ
    float* topks    = (float*)(sortidx + (size_t)nimg * SORT_SZ);    // N*KTOP
    int*   topki    = (int*)(topks + (size_t)nimg * KTOP);           // N*KTOP

    int tot = nimg * SORT_SZ;
    decode_kernel<<<(tot + 255) / 256, 256, 0, stream>>>(
        cls0, cls1, cls2, reg0, reg1, reg2, obj0, obj1, obj2,
        boxes_ws, labels_ws, sortkey, sortidx, nimg);
    sort_kernel<<<nimg, 1024, 0, stream>>>(sortkey, sortidx, topks, topki);
    nms_kernel<<<nimg, 1024, 0, stream>>>(boxes_ws, labels_ws, topks, topki,
                                          (float*)d_out, nimg);
    (void)n_in; (void)out_size; (void)ws_size;
}